// EncoderFeatureLoss_78606491452003
// MI455X (gfx1250) — compile-verified
//
#include <hip/hip_runtime.h>
#include <hip/hip_bf16.h>

namespace {
constexpr int   kHW  = 16384;    // H*W = 128*128
constexpr int   kN   = 8;
constexpr int   kC   = 256;
constexpr int   kNB  = 20;
constexpr float kInvTemp = 2.0f;          // 1/0.5
constexpr float kInvHW   = 1.0f / 16384.0f;
constexpr float kInvC    = 1.0f / 256.0f;

// workspace layout (float offsets)
constexpr int OFF_ROWSUM  = 0;                      // N*HW : fea_map[n][hw]
constexpr int OFF_CHANSUM = kN * kHW;               // N*C  : sum_hw |T|
constexpr int OFF_SQRTC   = OFF_CHANSUM + kN * kC;  // N*C  : sqrt(C_att)
constexpr int OFF_ZS      = OFF_SQRTC + kN * kC;    // N    : spatial softmax denom
constexpr int OFF_BGCNT   = OFF_ZS + kN;            // N    : #bg cells (as float)
constexpr int OFF_COEF    = OFF_BGCNT + kN;         // N*HW : fg mask, then coef
}

typedef __attribute__((ext_vector_type(2))) float v2f;
typedef __attribute__((ext_vector_type(8))) float v8f;

__global__ __launch_bounds__(256) void k_init(float* __restrict__ ws,
                                              float* __restrict__ out) {
  const int t = threadIdx.x;
  for (int i = t; i < kN * kC; i += 256) ws[OFF_CHANSUM + i] = 0.0f;
  if (t < 2 * kN) ws[OFF_ZS + t] = 0.0f;   // zeros Zs + bgcnt (contiguous)
  if (t == 0) *out = 0.0f;
}

// One pass over preds_T: rowsum[n][hw] = mean_c |T|, chansum[n][c] = sum_hw |T|.
// Each wave owns 32 hw rows of a single image n (so channel partials stay private).
__global__ __launch_bounds__(256) void k_stats(const float* __restrict__ T,
                                               float* __restrict__ ws) {
  const int lane = threadIdx.x & 31;
  const int wv   = threadIdx.x >> 5;
  const int gw   = blockIdx.x * 8 + wv;     // 4096 waves total
  const int n    = gw & 7;
  const int hw0  = (gw >> 3) << 5;          // 32 rows per wave
  const float4* __restrict__ T4 = (const float4*)T;
  float* __restrict__ rowsum  = ws + OFF_ROWSUM;
  float* __restrict__ chansum = ws + OFF_CHANSUM;

  float acc[8] = {0, 0, 0, 0, 0, 0, 0, 0};
  for (int i = 0; i < 32; ++i) {
    const int hw = hw0 + i;
    const long r = (long)hw * kN + n;            // memory row (hw*N + n)
    const long b4 = r * (kC / 4) + lane * 2;     // lane covers 8 contiguous channels
    float4 a = T4[b4];
    float4 b = T4[b4 + 1];
    float v0 = fabsf(a.x), v1 = fabsf(a.y), v2 = fabsf(a.z), v3 = fabsf(a.w);
    float v4 = fabsf(b.x), v5 = fabsf(b.y), v6 = fabsf(b.z), v7 = fabsf(b.w);
    acc[0] += v0; acc[1] += v1; acc[2] += v2; acc[3] += v3;
    acc[4] += v4; acc[5] += v5; acc[6] += v6; acc[7] += v7;
    float rs = v0 + v1 + v2 + v3 + v4 + v5 + v6 + v7;
    #pragma unroll
    for (int off = 16; off > 0; off >>= 1) rs += __shfl_xor(rs, off, 32);
    if (lane == 0) rowsum[n * kHW + hw] = rs * kInvC;
  }
  #pragma unroll
  for (int j = 0; j < 8; ++j)
    atomicAdd(chansum + n * kC + lane * 8 + j, acc[j]);
}

// One block per image n: channel softmax -> sqrtC, spatial softmax denom -> Zs.
__global__ __launch_bounds__(256) void k_softmax(float* __restrict__ ws) {
  const int n = blockIdx.x;
  const int t = threadIdx.x;
  __shared__ float red[8];
  __shared__ float zc_s;

  // channel attention: C_att = C * softmax(chansum/HW / temp)
  const float e = expf(ws[OFF_CHANSUM + n * kC + t] * kInvHW * kInvTemp);
  float v = e;
  #pragma unroll
  for (int off = 16; off > 0; off >>= 1) v += __shfl_xor(v, off, 32);
  if ((t & 31) == 0) red[t >> 5] = v;
  __syncthreads();
  if (t == 0) {
    float s = 0.0f;
    for (int i = 0; i < 8; ++i) s += red[i];
    zc_s = s;
  }
  __syncthreads();
  ws[OFF_SQRTC + n * kC + t] = sqrtf((float)kC * e / zc_s);

  // spatial attention denominator Zs[n] = sum_hw exp(fea_map/temp)
  const float* rowsum = ws + OFF_ROWSUM;
  float s = 0.0f;
  for (int hw = t; hw < kHW; hw += 256)
    s += expf(rowsum[n * kHW + hw] * kInvTemp);
  #pragma unroll
  for (int off = 16; off > 0; off >>= 1) s += __shfl_xor(s, off, 32);
  __syncthreads();
  if ((t & 31) == 0) red[t >> 5] = s;
  __syncthreads();
  if (t == 0) {
    float z = 0.0f;
    for (int i = 0; i < 8; ++i) z += red[i];
    ws[OFF_ZS + n] = z;
  }
}

// Foreground mask via max-scatter over 20 boxes; count bg cells via wave ballot.
__global__ __launch_bounds__(256) void k_mask(const float4* __restrict__ boxes,
                                              float* __restrict__ ws) {
  const int j = blockIdx.x * 256 + threadIdx.x;   // < N*HW
  const int n  = j >> 14;
  const int hw = j & (kHW - 1);
  const int h = hw >> 7, w = hw & 127;
  float fgv = 0.0f;
  for (int b = 0; b < kNB; ++b) {
    const float4 bb = boxes[n * kNB + b];         // (x1,y1,x2,y2)
    const int wmin = (int)floorf(bb.x / 800.0f * 128.0f);
    const int hmin = (int)floorf(bb.y / 800.0f * 128.0f);
    const int wmax = (int)ceilf (bb.z / 800.0f * 128.0f);
    const int hmax = (int)ceilf (bb.w / 800.0f * 128.0f);
    if (h >= hmin && h <= hmax && w >= wmin && w <= wmax) {
      const float area = 1.0f / (float)(hmax + 1 - hmin) / (float)(wmax + 1 - wmin);
      fgv = fmaxf(fgv, area);
    }
  }
  ws[OFF_COEF + j] = fgv;                         // stash fg mask
  const unsigned long long m = __ballot(fgv <= 0.0f);  // n uniform per wave
  if ((threadIdx.x & 31) == 0)
    atomicAdd(ws + OFF_BGCNT + n, (float)__popcll(m));
}

// coef[n][hw] = sqrt(S_att) * (sqrt(Mfg) + sqrt(Mbg))   (overwrites fg in place)
__global__ __launch_bounds__(256) void k_coef(float* __restrict__ ws) {
  const int j = blockIdx.x * 256 + threadIdx.x;
  const int n = j >> 14;
  const float satt = (float)kHW * expf(ws[OFF_ROWSUM + j] * kInvTemp) / ws[OFF_ZS + n];
  const float fgv  = ws[OFF_COEF + j];
  const float sqbg = (fgv <= 0.0f) ? rsqrtf(ws[OFF_BGCNT + n]) : 0.0f;
  ws[OFF_COEF + j] = sqrtf(satt) * (sqrtf(fgv) + sqbg);
}

// Main reduction: loss += (1/HW) * sum_rows coef[row] * sum_c sqrtC[n][c]*|S-T|.
// Block = 256 thr (8 waves), 128 memory rows per block, 64-channel LDS chunks.
// Per wave: 16 rows as the A-matrix of chained V_WMMA_F32_16X16X4_F32, B = ones.
__global__ __launch_bounds__(256) void k_main(const float* __restrict__ S,
                                              const float* __restrict__ T,
                                              const float* __restrict__ ws,
                                              float* __restrict__ out) {
  __shared__ float sd[128 * 68];                  // stride 68: bank-conflict free
  const int tid  = threadIdx.x;
  const int lane = tid & 31;
  const int wv   = tid >> 5;
  const int rowbase = blockIdx.x * 128;
  const float4* __restrict__ S4   = (const float4*)S;
  const float4* __restrict__ T4   = (const float4*)T;
  const float4* __restrict__ sqc4 = (const float4*)(ws + OFF_SQRTC);
  const float*  __restrict__ coef = ws + OFF_COEF;

  v8f acc = {0, 0, 0, 0, 0, 0, 0, 0};
  const v2f ones = {1.0f, 1.0f};
  // A-matrix layout (16x4 f32): lanes 0-15 supply K=0,1 of row M=lane;
  // lanes 16-31 supply K=2,3 of row M=lane-16.
  const int rloc  = wv * 16 + (lane & 15);
  const int chsel = (lane >> 4) << 1;             // 0 or 2

  for (int ch0 = 0; ch0 < kC; ch0 += 64) {
    // stage 128 rows x 64 channels of coef*sqrtC*|S-T| into LDS (float4 coalesced)
    #pragma unroll
    for (int it = 0; it < 8; ++it) {
      const int idx = tid + it * 256;             // 2048 float4 elements
      const int r  = idx >> 4;
      const int c4 = (idx & 15) << 2;
      const int rg = rowbase + r;                 // memory row = hw*N + n
      const int n  = rg & 7;
      const int hw = rg >> 3;
      const long g4 = (long)rg * (kC / 4) + ((ch0 + c4) >> 2);
      const float4 sv = S4[g4];
      const float4 tv = T4[g4];
      const float4 q  = sqc4[n * (kC / 4) + ((ch0 + c4) >> 2)];
      const float  cf = coef[n * kHW + hw];
      float4 d;
      d.x = fabsf(sv.x - tv.x) * q.x * cf;
      d.y = fabsf(sv.y - tv.y) * q.y * cf;
      d.z = fabsf(sv.z - tv.z) * q.z * cf;
      d.w = fabsf(sv.w - tv.w) * q.w * cf;
      *(float4*)&sd[r * 68 + c4] = d;
    }
    __syncthreads();
    // 16 chained-accumulate WMMAs cover this 64-channel chunk (K=4 each)
    #pragma unroll
    for (int k0 = 0; k0 < 64; k0 += 4) {
      const v2f a = *(const v2f*)&sd[rloc * 68 + k0 + chsel];
      acc = __builtin_amdgcn_wmma_f32_16x16x4_f32(
          /*neg_a=*/false, a, /*neg_b=*/false, ones,
          /*c_mod=*/(short)0, acc, /*reuse_a=*/false, /*reuse_b=*/false);
    }
    __syncthreads();
  }

  // D[i][j] identical over j (B columns identical). Lanes 0-15 hold tile rows
  // 0-7 in acc[0..7]; lanes 16-31 hold rows 8-15. Sum both halves via shfl.
  float t8 = acc[0] + acc[1] + acc[2] + acc[3] + acc[4] + acc[5] + acc[6] + acc[7];
  const float lo = __shfl(t8, 0, 32);
  const float hi = __shfl(t8, 16, 32);
  if (lane == 0) atomicAdd(out, (lo + hi) * kInvHW);
}

extern "C" void kernel_launch(void* const* d_in, const int* in_sizes, int n_in,
                              void* d_out, int out_size, void* d_ws, size_t ws_size,
                              hipStream_t stream) {
  (void)in_sizes; (void)n_in; (void)out_size; (void)ws_size;
  const float*  S     = (const float*)d_in[0];   // preds_S (HW, N, C)
  const float*  T     = (const float*)d_in[1];   // preds_T (HW, N, C)
  const float4* boxes = (const float4*)d_in[2];  // gt_bboxes (N, NB, 4)
  float* out = (float*)d_out;
  float* ws  = (float*)d_ws;

  k_init   <<<1,    256, 0, stream>>>(ws, out);
  k_stats  <<<512,  256, 0, stream>>>(T, ws);
  k_softmax<<<kN,   256, 0, stream>>>(ws);
  k_mask   <<<512,  256, 0, stream>>>(boxes, ws);
  k_coef   <<<512,  256, 0, stream>>>(ws);
  k_main   <<<1024, 256, 0, stream>>>(S, T, ws, out);
}